// FullyFrameAttention_70042326663495
// MI455X (gfx1250) — compile-verified
//
#include <hip/hip_runtime.h>

typedef __attribute__((ext_vector_type(2))) float v2f;
typedef __attribute__((ext_vector_type(8))) float v8f;
typedef __attribute__((ext_vector_type(2))) int   i2v;
typedef __attribute__((ext_vector_type(4))) int   i4v;

#define S_TOK   4096
#define CDIM    512
#define NH      8
#define HD      64
#define QKV_LD  1536           // row stride of flat qkv buffer
#define LOG2E   1.4426950408889634f

// ---------------------------------------------------------------------------
// gfx1250 async global->LDS path (GLOBAL_LOAD_ASYNC_TO_LDS_*, ASYNCcnt),
// guarded so the kernel still builds if the toolchain lacks the builtins.
// Builtin signature (from clang diagnostic): (vNi addrspace(1)*, vNi addrspace(3)*, Imm, Imm)
// ---------------------------------------------------------------------------
#ifndef USE_ASYNC_LDS
#if defined(__has_builtin)
#if __has_builtin(__builtin_amdgcn_global_load_async_to_lds_b64) && \
    __has_builtin(__builtin_amdgcn_global_load_async_to_lds_b128)
#define USE_ASYNC_LDS 1
#endif
#endif
#endif
#ifndef USE_ASYNC_LDS
#define USE_ASYNC_LDS 0
#endif

#define AS1 __attribute__((address_space(1)))
#define AS3 __attribute__((address_space(3)))

#if USE_ASYNC_LDS
static __device__ __forceinline__ void cp_async_b64(const void* g, void* l) {
  __builtin_amdgcn_global_load_async_to_lds_b64((AS1 i2v*)g, (AS3 i2v*)l, 0, 0);
}
static __device__ __forceinline__ void cp_async_b128(const void* g, void* l) {
  __builtin_amdgcn_global_load_async_to_lds_b128((AS1 i4v*)g, (AS3 i4v*)l, 0, 0);
}
static __device__ __forceinline__ void async_wait_all() {
#if defined(__has_builtin) && __has_builtin(__builtin_amdgcn_s_wait_asynccnt)
  __builtin_amdgcn_s_wait_asynccnt(0);
#else
  asm volatile("s_wait_asynccnt 0x0" ::: "memory");
#endif
}
#else
static __device__ __forceinline__ void async_wait_all() {}
#endif

static __device__ __forceinline__ v8f wmma_f32(v2f a, v2f b, v8f c) {
  return __builtin_amdgcn_wmma_f32_16x16x4_f32(false, a, false, b, (short)0, c,
                                               false, false);
}

// ===========================================================================
// Generic fp32 WMMA GEMM: C[M,N] = A[M,K] * B[K,N] (+ bias[N])
// 256 threads (8 waves). WG tile 256(M) x 64(N); wave tile 32x64.
// K staged by 16 in WMMA-fragment-ready LDS layout, double buffered.
//   A frags: group g = (k_local>>1) in 0..7, stride 272 float2, element [row]
//   B frags: group g likewise, stride 80 float2, element [col]
// ===========================================================================
#define AG 272   // A group stride (float2): 256 rows + 16 pad (half-disjoint banks)
#define BG 80    // B group stride (float2): 64 cols + 16 pad

static __device__ __forceinline__ void stage_a(float* abuf, const float* A,
                                               int lda, int m0, int k0, int tid) {
  const float* src = A + (size_t)(m0 + tid) * lda + k0;
#if USE_ASYNC_LDS
#pragma unroll
  for (int g = 0; g < 8; ++g)
    cp_async_b64(src + 2 * g, &abuf[(g * AG + tid) * 2]);
#else
#pragma unroll
  for (int g = 0; g < 8; ++g) {
    v2f t;
    t.x = src[2 * g];
    t.y = src[2 * g + 1];
    *(v2f*)&abuf[(g * AG + tid) * 2] = t;
  }
#endif
}

static __device__ __forceinline__ void stage_b(float* bbuf, const float* Bw,
                                               int N, int n0, int k0, int tid) {
  const int g = tid >> 5, n2 = (tid & 31) * 2;   // interleave two K rows
  const float* s0 = Bw + (size_t)(k0 + 2 * g) * N + n0 + n2;
  const v2f r0 = *(const v2f*)s0;
  const v2f r1 = *(const v2f*)(s0 + N);
  const float4 w = {r0.x, r1.x, r0.y, r1.y};
  *(float4*)&bbuf[(g * BG + n2) * 2] = w;
}

__global__ __launch_bounds__(256) void wmma_gemm_f32(
    const float* __restrict__ A, const float* __restrict__ Bw,
    const float* __restrict__ bias, float* __restrict__ Cout,
    int M, int N, int Kd) {
  __shared__ float aF[2][8 * AG * 2];
  __shared__ float bF[2][8 * BG * 2];

  const int tid  = threadIdx.x;
  const int wave = tid >> 5;
  const int lane = tid & 31;
  const int l16  = lane & 15;
  const int half = lane >> 4;

  const int m0 = blockIdx.y * 256;
  const int n0 = blockIdx.x * 64;
  (void)M;

  v8f a00 = {}, a01 = {}, a02 = {}, a03 = {};
  v8f a10 = {}, a11 = {}, a12 = {}, a13 = {};

  stage_a(aF[0], A, Kd, m0, 0, tid);
  stage_b(bF[0], Bw, N, n0, 0, tid);

  const int rA = wave * 32 + l16;
  int buf = 0;
  for (int k0 = 0; k0 < Kd; k0 += 16) {
    async_wait_all();
    __syncthreads();
    if (k0 + 16 < Kd) {
      stage_a(aF[buf ^ 1], A, Kd, m0, k0 + 16, tid);
      stage_b(bF[buf ^ 1], Bw, N, n0, k0 + 16, tid);
    }
    const float* ab = aF[buf];
    const float* bb = bF[buf];
#pragma unroll
    for (int kk = 0; kk < 4; ++kk) {
      const int g = 2 * kk + half;             // per-half K pair (ISA layout)
      const v2f fa0 = *(const v2f*)&ab[(g * AG + rA) * 2];
      const v2f fa1 = *(const v2f*)&ab[(g * AG + rA + 16) * 2];
      const v2f fb0 = *(const v2f*)&bb[(g * BG + l16) * 2];
      const v2f fb1 = *(const v2f*)&bb[(g * BG + l16 + 16) * 2];
      const v2f fb2 = *(const v2f*)&bb[(g * BG + l16 + 32) * 2];
      const v2f fb3 = *(const v2f*)&bb[(g * BG + l16 + 48) * 2];
      a00 = wmma_f32(fa0, fb0, a00);
      a01 = wmma_f32(fa0, fb1, a01);
      a02 = wmma_f32(fa0, fb2, a02);
      a03 = wmma_f32(fa0, fb3, a03);
      a10 = wmma_f32(fa1, fb0, a10);
      a11 = wmma_f32(fa1, fb1, a11);
      a12 = wmma_f32(fa1, fb2, a12);
      a13 = wmma_f32(fa1, fb3, a13);
    }
    buf ^= 1;
  }

  const float bb0 = bias ? bias[n0      + l16] : 0.0f;
  const float bb1 = bias ? bias[n0 + 16 + l16] : 0.0f;
  const float bb2 = bias ? bias[n0 + 32 + l16] : 0.0f;
  const float bb3 = bias ? bias[n0 + 48 + l16] : 0.0f;
  const int mb = m0 + wave * 32;
#pragma unroll
  for (int v = 0; v < 8; ++v) {
    const int r0 = mb + v + 8 * half;
    float* d0 = Cout + (size_t)r0 * N + n0 + l16;
    d0[0]  = a00[v] + bb0;
    d0[16] = a01[v] + bb1;
    d0[32] = a02[v] + bb2;
    d0[48] = a03[v] + bb3;
    float* d1 = d0 + (size_t)16 * N;
    d1[0]  = a10[v] + bb0;
    d1[16] = a11[v] + bb1;
    d1[32] = a12[v] + bb2;
    d1[48] = a13[v] + bb3;
  }
}

// ===========================================================================
// Flash attention: one (head, 128-query block) per WG; 8 waves, each wave
// owns 16 queries x hd=64. K tile staged fragment-ready (b64 frags), V tile
// raw row-major (pad 68), P transposed via per-wave column-major LDS tile.
// Double-buffered K/V with async global->LDS.
// ===========================================================================
#define PCS 20   // P tile column stride (floats)

static __device__ __forceinline__ void stage_kv(float* kbuf, float* vbuf,
                                                const float* Kh, const float* Vh,
                                                int kt, int tid) {
  const int r = tid >> 4, c = (tid & 15) * 4;
  const float* ks = Kh + (size_t)(kt * 16 + r) * QKV_LD + c;
  const float* vs = Vh + (size_t)(kt * 16 + r) * QKV_LD + c;
#if USE_ASYNC_LDS
  cp_async_b64(ks,     &kbuf[(((c >> 1)    ) * 16 + r) * 2]);
  cp_async_b64(ks + 2, &kbuf[(((c >> 1) + 1) * 16 + r) * 2]);
  cp_async_b128(vs, &vbuf[r * 68 + c]);
#else
  const float4 kv = *(const float4*)ks;
  v2f t0; t0.x = kv.x; t0.y = kv.y;
  v2f t1; t1.x = kv.z; t1.y = kv.w;
  *(v2f*)&kbuf[(((c >> 1)    ) * 16 + r) * 2] = t0;
  *(v2f*)&kbuf[(((c >> 1) + 1) * 16 + r) * 2] = t1;
  *(float4*)&vbuf[r * 68 + c] = *(const float4*)vs;
#endif
}

__global__ __launch_bounds__(256) void wmma_flash_attn(
    const float* __restrict__ qkv, float* __restrict__ O) {
  __shared__ float kF[2][32 * 16 * 2];   // fragment-ready K (float2 grid)
  __shared__ float vT[2][16 * 68];       // raw V rows, padded
  __shared__ float pT[8][16 * PCS];      // per-wave P, column-major

  const int tid  = threadIdx.x;
  const int wave = tid >> 5;
  const int lane = tid & 31;
  const int l16  = lane & 15;
  const int half = lane >> 4;

  const int head = blockIdx.y;
  const int q0   = blockIdx.x * 128 + wave * 16;

  const float* Qh = qkv + head * HD;
  const float* Kh = qkv + CDIM + head * HD;
  const float* Vh = qkv + 2 * CDIM + head * HD;

  // preload Q A-fragments (16 k-steps over hd=64) as b64 loads
  v2f qf[16];
#pragma unroll
  for (int i = 0; i < 16; ++i)
    qf[i] = *(const v2f*)(Qh + (size_t)(q0 + l16) * QKV_LD + 4 * i + 2 * half);

  v8f o0 = {}, o1 = {}, o2 = {}, o3 = {};
  float rm[8], rl[8];
#pragma unroll
  for (int v = 0; v < 8; ++v) { rm[v] = -1e30f; rl[v] = 0.0f; }

  const float kscale = 0.125f * LOG2E;
  float* pw = &pT[wave][0];

  stage_kv(kF[0], vT[0], Kh, Vh, 0, tid);
  int buf = 0;
  for (int kt = 0; kt < S_TOK / 16; ++kt) {
    async_wait_all();
    __syncthreads();
    if (kt + 1 < S_TOK / 16)
      stage_kv(kF[buf ^ 1], vT[buf ^ 1], Kh, Vh, kt + 1, tid);
    const float* kbuf = kF[buf];
    const float* vbuf = vT[buf];

    // scores = Q * K^T : B-frags are single b64 loads, conflict-free
    v8f s = {};
#pragma unroll
    for (int i = 0; i < 16; ++i) {
      const v2f b = *(const v2f*)&kbuf[((2 * i + half) * 16 + l16) * 2];
      s = wmma_f32(qf[i], b, s);
    }

    // online softmax (row = vgpr idx + 8*half; cols across 16-lane halves)
    float p[8];
#pragma unroll
    for (int v = 0; v < 8; ++v) {
      const float ss = s[v] * kscale;
      float mx = ss;
      mx = fmaxf(mx, __shfl_xor(mx, 1));
      mx = fmaxf(mx, __shfl_xor(mx, 2));
      mx = fmaxf(mx, __shfl_xor(mx, 4));
      mx = fmaxf(mx, __shfl_xor(mx, 8));
      const float mn   = fmaxf(rm[v], mx);
      const float pe   = __builtin_amdgcn_exp2f(ss - mn);
      const float corr = __builtin_amdgcn_exp2f(rm[v] - mn);
      rm[v] = mn;
      float rs = pe;
      rs += __shfl_xor(rs, 1);
      rs += __shfl_xor(rs, 2);
      rs += __shfl_xor(rs, 4);
      rs += __shfl_xor(rs, 8);
      rl[v] = rl[v] * corr + rs;
      o0[v] *= corr; o1[v] *= corr; o2[v] *= corr; o3[v] *= corr;
      p[v] = pe;
    }

    // P (C-layout) -> column-major LDS tile: two b128 stores per lane
    {
      const float4 lo = {p[0], p[1], p[2], p[3]};
      const float4 hi = {p[4], p[5], p[6], p[7]};
      float* dst = &pw[l16 * PCS + 8 * half];
      *(float4*)dst = lo;
      *(float4*)(dst + 4) = hi;
    }
    // same-wave LDS ops are in-order; loads below see the stores

    // O += P * V
#pragma unroll
    for (int kk = 0; kk < 4; ++kk) {
      const int c = 4 * kk + 2 * half;
      v2f a;
      a.x = pw[c * PCS + l16];
      a.y = pw[(c + 1) * PCS + l16];
      const int rb = c * 68;
      v2f b0, b1, b2, b3;
      b0.x = vbuf[rb      + l16];  b0.y = vbuf[rb + 68      + l16];
      b1.x = vbuf[rb + 16 + l16];  b1.y = vbuf[rb + 68 + 16 + l16];
      b2.x = vbuf[rb + 32 + l16];  b2.y = vbuf[rb + 68 + 32 + l16];
      b3.x = vbuf[rb + 48 + l16];  b3.y = vbuf[rb + 68 + 48 + l16];
      o0 = wmma_f32(a, b0, o0);
      o1 = wmma_f32(a, b1, o1);
      o2 = wmma_f32(a, b2, o2);
      o3 = wmma_f32(a, b3, o3);
    }
    buf ^= 1;
  }

  // epilogue: O[q][head*64 + d] = acc / rowsum
#pragma unroll
  for (int v = 0; v < 8; ++v) {
    const int q = q0 + v + 8 * half;
    const float inv = 1.0f / rl[v];
    float* dst = O + (size_t)q * CDIM + head * HD + l16;
    dst[0]  = o0[v] * inv;
    dst[16] = o1[v] * inv;
    dst[32] = o2[v] * inv;
    dst[48] = o3[v] * inv;
  }
}

// ---------------------------------------------------------------------------
extern "C" void kernel_launch(void* const* d_in, const int* in_sizes, int n_in,
                              void* d_out, int out_size, void* d_ws,
                              size_t ws_size, hipStream_t stream) {
  (void)in_sizes; (void)n_in; (void)out_size; (void)ws_size;
  const float* x      = (const float*)d_in[0];   // (4096, 512)
  const float* w_qkv  = (const float*)d_in[1];   // (512, 1536)
  const float* w_proj = (const float*)d_in[2];   // (512, 512)
  const float* b_proj = (const float*)d_in[3];   // (512,)
  float* out = (float*)d_out;                    // (4096, 512)

  float* qkv  = (float*)d_ws;                    // (4096, 1536)
  float* attn = qkv + (size_t)S_TOK * QKV_LD;    // (4096, 512)

  dim3 blk(256);
  // 1) QKV projection: (4096x512) @ (512x1536)
  wmma_gemm_f32<<<dim3(QKV_LD / 64, S_TOK / 256), blk, 0, stream>>>(
      x, w_qkv, nullptr, qkv, S_TOK, QKV_LD, CDIM);
  // 2) Flash attention per head
  wmma_flash_attn<<<dim3(S_TOK / 128, NH), blk, 0, stream>>>(qkv, attn);
  // 3) Output projection: (4096x512) @ (512x512) + bias
  wmma_gemm_f32<<<dim3(CDIM / 64, S_TOK / 256), blk, 0, stream>>>(
      attn, w_proj, b_proj, out, S_TOK, CDIM, CDIM);
}